// DepGCN_214748365178
// MI455X (gfx1250) — compile-verified
//
#include <hip/hip_runtime.h>
#include <hip/hip_bf16.h>

// Problem constants (from reference)
#define DEP_NUM 50
#define DEP_DIM 64
#define IN_F    256
#define OUT_F   256
#define BB      8
#define LL      256
#define M_ROWS  (BB * LL)   // 2048

typedef float v2f __attribute__((ext_vector_type(2)));
typedef float v8f __attribute__((ext_vector_type(8)));

// ---------------------------------------------------------------------------
// Kernel 1: emb_dot[label] = dep_emb[label] . colsum(W_attn[IN_F:])
// Softmax is invariant to the per-row-constant terms (text.wa_text, sum(b_attn)),
// so this 50-entry LUT is the only piece of W_attn that affects the output.
// ---------------------------------------------------------------------------
__global__ void __launch_bounds__(64)
dep_prep_kernel(const float* __restrict__ W_attn,   // [IN_F+DEP_DIM, OUT_F]
                const float* __restrict__ dep_emb,  // [DEP_NUM, DEP_DIM]
                float* __restrict__ emb_dot)        // [64] (padded)
{
    __shared__ float s_wadep[DEP_DIM];
    const int t = threadIdx.x;            // 0..63
    const float* rowp = W_attn + (size_t)(IN_F + t) * OUT_F;
    float s = 0.0f;
    for (int o = 0; o < OUT_F; ++o) s += rowp[o];
    s_wadep[t] = s;
    __syncthreads();
    float d = 0.0f;
    if (t < DEP_NUM) {
        const float* e = dep_emb + (size_t)t * DEP_DIM;
        for (int k = 0; k < DEP_DIM; ++k) d += e[k] * s_wadep[k];
    }
    emb_dot[t] = d;                       // pad entries (50..63) written as 0
}

// ---------------------------------------------------------------------------
// Kernel 2: per (b,i) row: masked softmax over j, then
//   agg[row,k] = sum_j p_j * dep_emb[label_j, k]
// One block per row, 256 threads. Deterministic (no float atomics): the
// j-sum is split into 4 chunks of 64 and combined in a fixed order.
// ---------------------------------------------------------------------------
__global__ void __launch_bounds__(256)
row_softmax_agg_kernel(const int*   __restrict__ dep_mat,     // [M_ROWS, LL]
                       const int*   __restrict__ dep_labels,  // [M_ROWS, LL]
                       const float* __restrict__ dep_emb,     // [DEP_NUM, DEP_DIM]
                       const float* __restrict__ emb_dot,     // [64]
                       float*       __restrict__ agg)         // [M_ROWS, DEP_DIM]
{
    __shared__ float s_embdot[64];
    __shared__ float s_emb[DEP_NUM * DEP_DIM];  // 12.5 KB, LDS-resident table
    __shared__ float s_p[LL];
    __shared__ int   s_lab[LL];
    __shared__ float s_red[LL];
    __shared__ float s_part[LL];                // [4 chunks][64 k]

    const int tid = threadIdx.x;
    const int row = blockIdx.x;                 // 0..M_ROWS-1

    if (tid < 64) s_embdot[tid] = emb_dot[tid];
    for (int i = tid; i < DEP_NUM * DEP_DIM; i += 256) s_emb[i] = dep_emb[i];

    const int lab = dep_labels[(size_t)row * LL + tid];
    const int msk = dep_mat[(size_t)row * LL + tid];
    s_lab[tid] = lab;
    __syncthreads();

    float logit = (msk != 0) ? s_embdot[lab] : -1.0e30f;

    // block max
    s_red[tid] = logit; __syncthreads();
    for (int s = 128; s > 0; s >>= 1) {
        if (tid < s) { float o = s_red[tid + s]; if (o > s_red[tid]) s_red[tid] = o; }
        __syncthreads();
    }
    const float mx = s_red[0];
    __syncthreads();

    const float ev = __expf(logit - mx);

    // block sum
    s_red[tid] = ev; __syncthreads();
    for (int s = 128; s > 0; s >>= 1) {
        if (tid < s) s_red[tid] += s_red[tid + s];
        __syncthreads();
    }
    const float inv = 1.0f / s_red[0];
    s_p[tid] = ev * inv;
    __syncthreads();

    // agg: thread (chunk=tid>>6, k=tid&63) handles 64 j's; combine 4 partials
    const int k  = tid & 63;
    const int ch = tid >> 6;
    const int j0 = ch * 64;
    float acc = 0.0f;
    #pragma unroll 4
    for (int j = 0; j < 64; ++j) {
        const int jj = j0 + j;
        acc += s_p[jj] * s_emb[s_lab[jj] * DEP_DIM + k];
    }
    s_part[ch * 64 + k] = acc;
    __syncthreads();
    if (tid < 64) {
        agg[(size_t)row * DEP_DIM + tid] =
            s_part[tid] + s_part[64 + tid] + s_part[128 + tid] + s_part[192 + tid];
    }
}

// ---------------------------------------------------------------------------
// Kernel 3: out = relu(text + agg @ W_fc + b_fc)
// GEMM M=2048, K=64, N=256 in pure f32 via V_WMMA_F32_16X16X4_F32.
// One wave per 16x16 tile, 8 waves/block, 256 blocks (2048 tiles).
// VGPR layouts per CDNA5 ISA 7.12.2:
//   A 16x4:  lanes 0-15 -> {K=0,K=1}, lanes 16-31 -> {K=2,K=3}
//   B 4x16:  lanes 0-15 -> rows K=0,1; lanes 16-31 -> rows K=2,3 (N = lane&15)
//   C/D:     VGPR r: lanes 0-15 -> M=r, lanes 16-31 -> M=r+8
// ---------------------------------------------------------------------------
__global__ void __launch_bounds__(256)
wmma_gemm_out_kernel(const float* __restrict__ agg,   // [M_ROWS, DEP_DIM]
                     const float* __restrict__ W_fc,  // [DEP_DIM, OUT_F]
                     const float* __restrict__ text,  // [M_ROWS, OUT_F]
                     const float* __restrict__ b_fc,  // [OUT_F]
                     float*       __restrict__ out)   // [M_ROWS, OUT_F]
{
    const int tid  = threadIdx.x;
    const int wave = tid >> 5;
    const int lane = tid & 31;
    const int tile = blockIdx.x * 8 + wave;     // 0..2047
    const int tileN = tile & 15;                // 16 N-tiles
    const int tileM = tile >> 4;                // 128 M-tiles

    const int half = lane >> 4;                 // 0: K lo pair, 1: K hi pair
    const int l16  = lane & 15;

    const int m_row = tileM * 16 + l16;
    const int n_col = tileN * 16 + l16;
    const float* __restrict__ arow = agg + (size_t)m_row * DEP_DIM;

    v8f acc = {0.f, 0.f, 0.f, 0.f, 0.f, 0.f, 0.f, 0.f};

    #pragma unroll
    for (int k0 = 0; k0 < DEP_DIM; k0 += 4) {
        const int kb = k0 + half * 2;
        v2f a, b;
        a.x = arow[kb];
        a.y = arow[kb + 1];
        b.x = W_fc[(size_t)kb * OUT_F + n_col];
        b.y = W_fc[(size_t)(kb + 1) * OUT_F + n_col];
        acc = __builtin_amdgcn_wmma_f32_16x16x4_f32(
                  /*neg_a=*/false, a, /*neg_b=*/false, b,
                  /*c_mod=*/(short)0, acc, /*reuse_a=*/false, /*reuse_b=*/false);
    }

    const float bias = b_fc[n_col];
    #pragma unroll
    for (int r = 0; r < 8; ++r) {
        const int row = tileM * 16 + half * 8 + r;
        const size_t idx = (size_t)row * OUT_F + n_col;
        const float v = text[idx] + acc[r] + bias;
        out[idx] = v > 0.0f ? v : 0.0f;
    }
}

// ---------------------------------------------------------------------------
// Launcher
// ---------------------------------------------------------------------------
extern "C" void kernel_launch(void* const* d_in, const int* in_sizes, int n_in,
                              void* d_out, int out_size, void* d_ws, size_t ws_size,
                              hipStream_t stream) {
    const float* text       = (const float*)d_in[0];
    const int*   dep_mat    = (const int*)  d_in[1];
    const int*   dep_labels = (const int*)  d_in[2];
    const float* dep_emb    = (const float*)d_in[3];
    const float* W_attn     = (const float*)d_in[4];
    // d_in[5] = b_attn: cancels in the softmax, provably unused.
    const float* W_fc       = (const float*)d_in[6];
    const float* b_fc       = (const float*)d_in[7];
    float* out = (float*)d_out;

    // Workspace layout: emb_dot[64] | agg[2048*64]
    float* emb_dot = (float*)d_ws;
    float* agg     = emb_dot + 64;

    dep_prep_kernel<<<1, 64, 0, stream>>>(W_attn, dep_emb, emb_dot);
    row_softmax_agg_kernel<<<M_ROWS, 256, 0, stream>>>(dep_mat, dep_labels,
                                                       dep_emb, emb_dot, agg);
    wmma_gemm_out_kernel<<<M_ROWS / 8, 256, 0, stream>>>(agg, W_fc, text, b_fc, out);
}